// RocketKVCache_39041252720707
// MI455X (gfx1250) — compile-verified
//
#include <hip/hip_runtime.h>
#include <hip/hip_bf16.h>
#include <stdint.h>

// Problem dims (fixed by reference)
#define BB    4
#define HQn   32
#define HKn   8
#define Sn    8192
#define Dn    128
#define Wn    32
#define KPOOL 63
#define Pn    1024
#define Cn    16
#define Gn    (HQn / HKn)   // 4
#define SWn   (Sn - Wn)     // 8160
#define PWn   (Pn - Wn)     // 992
#define QLD   132           // padded LDS row pitch (floats): 132 mod 64 = 4 -> conflict-free
#define NPART 4
#define IPP   16            // 64 key-tiles / NPART

#define FINF __builtin_huge_valf()

typedef float v2f __attribute__((ext_vector_type(2)));
typedef float v4f __attribute__((ext_vector_type(4)));
typedef float v8f __attribute__((ext_vector_type(8)));

static __device__ __forceinline__ v8f rkv_wmma4(v2f a, v2f b, v8f c) {
  // D(16x16,f32) = A(16x4,f32) x B(4x16,f32) + C ; exact fp32
  return __builtin_amdgcn_wmma_f32_16x16x4_f32(false, a, false, b, (short)0, c, false, false);
}

static __device__ __forceinline__ void rkv_load_q(const float* __restrict__ qg, float* qlds) {
  for (int i = threadIdx.x; i < Wn * Dn; i += 256) {
    int w = i >> 7, d = i & 127;
    qlds[w * QLD + d] = qg[w * Dn + d];
  }
}

// One 32(query) x 16(key) score tile, contraction over D=128 via 32 WMMA_F32_16X16X4_F32.
// k-slot remap: WMMA pair (2j,2j+1) contracts d in [8j, 8j+8); lane group hi covers
// d = 8j+4hi .. 8j+4hi+3 with ONE b128 load feeding both WMMAs:
//   wmma 2j   : slots {2hi,2hi+1} <-> d = 8j+4hi+{0,1}  (.xy)
//   wmma 2j+1 : slots {2hi,2hi+1} <-> d = 8j+4hi+{2,3}  (.zw)
// A (from Q, rows t / 16+t) uses the identical mapping, so the dot products match.
// All 16 B loads are issued up-front (distinct regs) for full memory-level parallelism.
static __device__ __forceinline__ void rkv_qk_tile(const float* __restrict__ krow,
                                                   const float* qlds, int t, int hi,
                                                   v8f& c0, v8f& c1) {
  v4f bv[16];
#pragma unroll
  for (int j = 0; j < 16; ++j)
    bv[j] = *(const v4f*)(krow + 8 * j + 4 * hi);
#pragma unroll
  for (int j = 0; j < 16; ++j) {
    const int d0 = 8 * j + 4 * hi;
    const v4f a0 = *(const v4f*)(qlds + t * QLD + d0);
    const v4f a1 = *(const v4f*)(qlds + (16 + t) * QLD + d0);
    const v2f b01 = {bv[j].x, bv[j].y};
    const v2f b23 = {bv[j].z, bv[j].w};
    const v2f a0lo = {a0.x, a0.y}, a0hi = {a0.z, a0.w};
    const v2f a1lo = {a1.x, a1.y}, a1hi = {a1.z, a1.w};
    c0 = rkv_wmma4(a0lo, b01, c0);
    c1 = rkv_wmma4(a1lo, b01, c1);
    c0 = rkv_wmma4(a0hi, b23, c0);
    c1 = rkv_wmma4(a1hi, b23, c1);
  }
}

// ---------------- Kernel 1a: per-partition softmax statistics (row max + sumexp) -------------
__global__ __launch_bounds__(256) void rkv_k1a(const float* __restrict__ q,
                                               const float* __restrict__ kv,
                                               float* __restrict__ partm,
                                               float* __restrict__ parts) {
  __shared__ float qlds[Wn * QLD];
  __shared__ float red_m[8][32];
  __shared__ float red_s[8][32];
  const int bq = blockIdx.x;          // b*HQ + hq
  const int part = blockIdx.y;
  const int b = bq / HQn, hq = bq % HQn, hk = hq / Gn;
  const float* kb = kv + ((size_t)(b * HKn + hk)) * Sn * Dn;
  rkv_load_q(q + (size_t)bq * Wn * Dn, qlds);
  __syncthreads();

  const int wave = threadIdx.x >> 5, lane = threadIdx.x & 31;
  const int t = lane & 15, hi = lane >> 4;
  const float scale = 0.088388347648318447f;  // 1/sqrt(128)

  float m_run[16], s_run[16];
#pragma unroll
  for (int i = 0; i < 16; ++i) { m_run[i] = -FINF; s_run[i] = 0.f; }

  for (int it = part * IPP; it < part * IPP + IPP; ++it) {
    const int n = it * 128 + wave * 16 + t;   // this lane's key index
    v8f c0 = {}; v8f c1 = {};
    rkv_qk_tile(kb + (size_t)n * Dn, qlds, t, hi, c0, c1);
#pragma unroll
    for (int mt = 0; mt < 2; ++mt) {
      v8f cc = mt ? c1 : c0;
#pragma unroll
      for (int r = 0; r < 8; ++r) {
        const int mrow = mt * 16 + r + 8 * hi;     // query row of this element
        float sc = cc[r] * scale;
        if (n > mrow + SWn) sc = -FINF;            // causal mask for last-W keys
        float tm = sc;                              // max over the 16 key-lanes of this group
        tm = fmaxf(tm, __shfl_xor(tm, 1));
        tm = fmaxf(tm, __shfl_xor(tm, 2));
        tm = fmaxf(tm, __shfl_xor(tm, 4));
        tm = fmaxf(tm, __shfl_xor(tm, 8));
        const int idx = mt * 8 + r;
        const float mold = m_run[idx];
        const float mnew = fmaxf(mold, tm);
        float e = (sc == -FINF) ? 0.f : __expf(sc - mnew);
        e += __shfl_xor(e, 1);
        e += __shfl_xor(e, 2);
        e += __shfl_xor(e, 4);
        e += __shfl_xor(e, 8);
        float srun = s_run[idx];
        if (mnew > mold) srun *= __expf(mold - mnew);
        s_run[idx] = srun + e;
        m_run[idx] = mnew;
      }
    }
  }
  if (t == 0) {
#pragma unroll
    for (int idx = 0; idx < 16; ++idx) {
      const int row = (idx >> 3) * 16 + (idx & 7) + 8 * hi;
      red_m[wave][row] = m_run[idx];
      red_s[wave][row] = s_run[idx];
    }
  }
  __syncthreads();
  if (threadIdx.x < 32) {
    float M = -FINF;
#pragma unroll
    for (int w2 = 0; w2 < 8; ++w2) M = fmaxf(M, red_m[w2][threadIdx.x]);
    float Z = 0.f;
#pragma unroll
    for (int w2 = 0; w2 < 8; ++w2) {
      const float mm = red_m[w2][threadIdx.x];
      if (mm != -FINF) Z += red_s[w2][threadIdx.x] * __expf(mm - M);
    }
    partm[((size_t)bq * NPART + part) * 32 + threadIdx.x] = M;
    parts[((size_t)bq * NPART + part) * 32 + threadIdx.x] = Z;
  }
}

// ---------------- Kernel 1b: normalized probs, summed over 32 window queries -----------------
__global__ __launch_bounds__(256) void rkv_k1b(const float* __restrict__ q,
                                               const float* __restrict__ kv,
                                               const float* __restrict__ partm,
                                               const float* __restrict__ parts,
                                               float* __restrict__ imp4) {
  __shared__ float qlds[Wn * QLD];
  __shared__ float Mrow[32];
  __shared__ float Zinv[32];
  const int bq = blockIdx.x;
  const int part = blockIdx.y;
  const int b = bq / HQn, hq = bq % HQn, hk = hq / Gn;
  const float* kb = kv + ((size_t)(b * HKn + hk)) * Sn * Dn;
  rkv_load_q(q + (size_t)bq * Wn * Dn, qlds);
  if (threadIdx.x < 32) {
    float M = -FINF;
#pragma unroll
    for (int p2 = 0; p2 < NPART; ++p2)
      M = fmaxf(M, partm[((size_t)bq * NPART + p2) * 32 + threadIdx.x]);
    float Z = 0.f;
#pragma unroll
    for (int p2 = 0; p2 < NPART; ++p2) {
      const float mm = partm[((size_t)bq * NPART + p2) * 32 + threadIdx.x];
      if (mm != -FINF) Z += parts[((size_t)bq * NPART + p2) * 32 + threadIdx.x] * __expf(mm - M);
    }
    Mrow[threadIdx.x] = M;
    Zinv[threadIdx.x] = 1.0f / Z;
  }
  __syncthreads();

  const int wave = threadIdx.x >> 5, lane = threadIdx.x & 31;
  const int t = lane & 15, hi = lane >> 4;
  const float scale = 0.088388347648318447f;
  float Ml[16], Zl[16];
#pragma unroll
  for (int idx = 0; idx < 16; ++idx) {
    const int row = (idx >> 3) * 16 + (idx & 7) + 8 * hi;
    Ml[idx] = Mrow[row];
    Zl[idx] = Zinv[row];
  }
  for (int it = part * IPP; it < part * IPP + IPP; ++it) {
    const int n = it * 128 + wave * 16 + t;
    v8f c0 = {}; v8f c1 = {};
    rkv_qk_tile(kb + (size_t)n * Dn, qlds, t, hi, c0, c1);
    float acc = 0.f;    // sum over this lane's 16 query rows
#pragma unroll
    for (int mt = 0; mt < 2; ++mt) {
      v8f cc = mt ? c1 : c0;
#pragma unroll
      for (int r = 0; r < 8; ++r) {
        const int idx = mt * 8 + r;
        acc += __expf(cc[r] * scale - Ml[idx]) * Zl[idx];
      }
    }
    acc += __shfl_xor(acc, 16);  // fold the two half-groups -> all 32 rows
    if (hi == 0 && n < SWn) imp4[(size_t)bq * Sn + n] = acc;
  }
}

// ---------------- Kernel 2: GQA group-sum + max_pool1d(k=63, pad=31) -------------------------
__global__ __launch_bounds__(256) void rkv_k2(const float* __restrict__ imp4,
                                              float* __restrict__ pooled) {
  __shared__ float buf[1024 + KPOOL - 1];
  const int bh = blockIdx.x >> 3;   // b*HK + hk
  const int chunk = blockIdx.x & 7;
  const int start = chunk * 1024;
  const int b = bh / HKn, hk = bh % HKn;
  const size_t base = ((size_t)(b * HQn + hk * Gn)) * Sn;
  for (int i = threadIdx.x; i < 1024 + KPOOL - 1; i += 256) {
    const int s = start - (KPOOL / 2) + i;
    float v = -FINF;
    if (s >= 0 && s < SWn) {
      v = 0.f;
#pragma unroll
      for (int g = 0; g < Gn; ++g) v += imp4[base + (size_t)g * Sn + s];
    }
    buf[i] = v;
  }
  __syncthreads();
  for (int o = threadIdx.x; o < 1024; o += 256) {
    const int s = start + o;
    if (s < SWn) {
      float mx = buf[o];
#pragma unroll
      for (int w2 = 1; w2 < KPOOL; ++w2) mx = fmaxf(mx, buf[o + w2]);
      pooled[(size_t)bh * Sn + s] = mx;
    }
  }
}

// ---------------- Kernel 3: exact top-k via bitonic sort in LDS ------------------------------
// key = (~ordered(val))<<32 | idx : ascending sort == descending value, ties -> lower idx first
__global__ __launch_bounds__(1024) void rkv_k3(const float* __restrict__ pooled,
                                               int* __restrict__ topk) {
  extern __shared__ unsigned long long skey[];   // 8192 * 8B = 64 KB LDS
  const int bh = blockIdx.x;
  for (int i = threadIdx.x; i < Sn; i += 1024) {
    unsigned long long key;
    if (i < SWn) {
      const float v = pooled[(size_t)bh * Sn + i];
      unsigned int u = __float_as_uint(v);
      u = (u & 0x80000000u) ? ~u : (u | 0x80000000u);   // monotonic float->uint
      key = (((unsigned long long)(~u)) << 32) | (unsigned int)i;
    } else {
      key = 0xFFFFFFFFFFFFFFFFULL;  // padding sorts last
    }
    skey[i] = key;
  }
  __syncthreads();
  for (int k = 2; k <= Sn; k <<= 1) {
    for (int j = k >> 1; j > 0; j >>= 1) {
      for (int i = threadIdx.x; i < Sn; i += 1024) {
        const int l = i ^ j;
        if (l > i) {
          const unsigned long long a = skey[i];
          const unsigned long long c = skey[l];
          const bool asc = ((i & k) == 0);
          if ((a > c) == asc) { skey[i] = c; skey[l] = a; }
        }
      }
      __syncthreads();
    }
  }
  for (int i = threadIdx.x; i < PWn; i += 1024)
    topk[(size_t)bh * PWn + i] = (int)(skey[i] & 0xFFFFFFFFULL);
}

// ---------------- Kernel 4: gather k_snap/v_snap + chunk min/max landmarks -------------------
__global__ __launch_bounds__(256) void rkv_k4(const float* __restrict__ kv_k,
                                              const float* __restrict__ kv_v,
                                              const int* __restrict__ topk,
                                              float* __restrict__ out) {
  const int bh = blockIdx.x;
  const size_t obase = (size_t)bh * (2 * Pn * Dn + 2 * Dn * (Pn / Cn));  // 278528
  const float* kb = kv_k + (size_t)bh * Sn * Dn;
  const float* vb = kv_v + (size_t)bh * Sn * Dn;
  const int* idx = topk + (size_t)bh * PWn;
  __shared__ int sidx[Pn];
  for (int i = threadIdx.x; i < Pn; i += 256)
    sidx[i] = (i < PWn) ? idx[i] : (SWn + i - PWn);   // append last W window rows
  __syncthreads();
  // k_snap / v_snap copies (float4 granularity)
  for (int task = threadIdx.x; task < Pn * (Dn / 4); task += 256) {
    const int row = task >> 5;
    const int quad = task & 31;
    const int src = sidx[row];
    const float4 kq = *(const float4*)(kb + (size_t)src * Dn + quad * 4);
    *(float4*)(out + obase + (size_t)row * Dn + quad * 4) = kq;
    const float4 vq = *(const float4*)(vb + (size_t)src * Dn + quad * 4);
    *(float4*)(out + obase + (size_t)Pn * Dn + (size_t)row * Dn + quad * 4) = vq;
  }
  // landmark cache kt: [2D, P/C]; first D rows = chunk min, next D = chunk max
  const size_t ktbase = obase + (size_t)2 * Pn * Dn;
  for (int task = threadIdx.x; task < (Pn / Cn) * Dn; task += 256) {
    const int j = task >> 7;
    const int d = task & 127;
    float mn = FINF, mx = -FINF;
#pragma unroll
    for (int c = 0; c < Cn; ++c) {
      const int src = sidx[j * Cn + c];
      const float val = kb[(size_t)src * Dn + d];
      mn = fminf(mn, val);
      mx = fmaxf(mx, val);
    }
    out[ktbase + (size_t)d * (Pn / Cn) + j] = mn;
    out[ktbase + (size_t)(Dn + d) * (Pn / Cn) + j] = mx;
  }
}

extern "C" void kernel_launch(void* const* d_in, const int* in_sizes, int n_in,
                              void* d_out, int out_size, void* d_ws, size_t ws_size,
                              hipStream_t stream) {
  const float* q  = (const float*)d_in[0];
  const float* kk = (const float*)d_in[1];
  const float* vv = (const float*)d_in[2];
  float* out = (float*)d_out;

  // workspace layout (floats), ~5.6 MB total
  float* imp4   = (float*)d_ws;                               // B*HQ*S
  float* pooled = imp4 + (size_t)BB * HQn * Sn;               // B*HK*S
  float* partm  = pooled + (size_t)BB * HKn * Sn;             // B*HQ*NPART*32
  float* parts  = partm + (size_t)BB * HQn * NPART * 32;      // B*HQ*NPART*32
  int*   topk   = (int*)(parts + (size_t)BB * HQn * NPART * 32);  // B*HK*PW

  dim3 g1(BB * HQn, NPART);
  rkv_k1a<<<g1, dim3(256), 0, stream>>>(q, kk, partm, parts);
  rkv_k1b<<<g1, dim3(256), 0, stream>>>(q, kk, partm, parts, imp4);
  rkv_k2<<<dim3(BB * HKn * 8), dim3(256), 0, stream>>>(imp4, pooled);
  rkv_k3<<<dim3(BB * HKn), dim3(1024), Sn * sizeof(unsigned long long), stream>>>(pooled, topk);
  rkv_k4<<<dim3(BB * HKn), dim3(256), 0, stream>>>(kk, vv, topk, out);
}